// xDeepFM_46273977647289
// MI455X (gfx1250) — compile-verified
//
#include <hip/hip_runtime.h>
#include <math.h>

typedef __attribute__((ext_vector_type(2))) float v2f;
typedef __attribute__((ext_vector_type(8))) float v8f;

#define M_   10
#define D_   10
#define H_   100
#define V_   200000
#define B_   12
#define NH   100      // DNN_H == H == 100
#define KPAD 112      // 100 padded to 7 * 16 columns

// One dense layer: Out(16x100) = relu( In(16x100) @ W(100x100)^T + b )
// computed as seven 16x16 tiles, K=100 in 25 steps of V_WMMA_F32_16X16X4_F32.
// Called uniformly by waves 0..6 (EXEC all-ones at the WMMA, as required).
__device__ __forceinline__ void dense_layer(const float (&In)[16][KPAD],
                                            const float* __restrict__ W,
                                            const float* __restrict__ bv,
                                            float (&Out)[16][KPAD],
                                            int wave, int lane)
{
    const int r    = lane & 15;        // A-matrix row / D-matrix column index
    const int koff = (lane >> 4) * 2;  // K sub-offset for the upper half-wave
    const int j    = wave * 16 + r;    // global output column
    const int jc   = (j < NH) ? j : (NH - 1);  // clamped (in-bounds) column

    v8f c = {};
    for (int k0 = 0; k0 < NH; k0 += 4) {
        v2f a, b;
        // A 16x4 fp32 layout: v0 = K=k0+koff, v1 = K=k0+koff+1, M = lane&15
        a.x = In[r][k0 + koff];
        a.y = In[r][k0 + koff + 1];
        // B 4x16 fp32 layout mirrors A: v0 = row K=k0+koff, v1 = K=k0+koff+1, N = lane&15
        // B[k][n] = W[j][k] (we need In @ W^T). Clamp load, zero via cndmask for j>=100.
        float bx = W[jc * NH + k0 + koff];
        float by = W[jc * NH + k0 + koff + 1];
        b.x = (j < NH) ? bx : 0.0f;
        b.y = (j < NH) ? by : 0.0f;
        c = __builtin_amdgcn_wmma_f32_16x16x4_f32(
                /*neg_a=*/false, a, /*neg_b=*/false, b,
                /*c_mod=*/(short)0, c, /*reuse_a=*/false, /*reuse_b=*/false);
    }
    const float bj = bv[jc];
    if (j < NH) {
        #pragma unroll
        for (int v = 0; v < 8; ++v) {
            // D layout: VGPR v holds M = v (lanes 0-15) / M = v+8 (lanes 16-31)
            int row = v + 8 * (lane >> 4);
            Out[row][j] = fmaxf(c[v] + bj, 0.0f);
        }
    }
}

__global__ __launch_bounds__(256, 1)
void xdeepfm_fused(const int*   __restrict__ x,
                   const float* __restrict__ emb,
                   const float* __restrict__ cin_w0,
                   const float* __restrict__ cin_w,
                   const float* __restrict__ W_linear,
                   const float* __restrict__ W_dnn,
                   const float* __restrict__ W_cin,
                   const float* __restrict__ bias,
                   const float* __restrict__ W1, const float* __restrict__ b1,
                   const float* __restrict__ W2, const float* __restrict__ b2,
                   const float* __restrict__ W3, const float* __restrict__ b3,
                   float* __restrict__ out)
{
    __shared__ float buf0[16][KPAD];
    __shared__ float buf1[16][KPAD];
    __shared__ float base[16];

    const int tid  = threadIdx.x;
    const int wave = tid >> 5;
    const int lane = tid & 31;

    // Warm L2/WGP$ for layer-2/3 weights while we gather embeddings.
    for (int i = tid; i < (NH * NH) / 16; i += 256) {
        __builtin_prefetch(W2 + i * 16, 0, 3);
        __builtin_prefetch(W3 + i * 16, 0, 3);
    }

    // Phase 1: h0 = X0.reshape(B, M*D) into buf0; pad rows>=12 / cols>=100 with 0.
    for (int i = tid; i < 16 * KPAD; i += 256) {
        int row = i / KPAD, col = i % KPAD;
        float v = 0.0f;
        if (row < B_ && col < NH) {
            int m = col / D_, d = col % D_;
            int e = x[row * M_ + m];
            v = emb[(m * V_ + e) * D_ + d];   // emb[m, x[b,m], d]
        }
        buf0[row][col] = v;
    }
    __syncthreads();

    if (wave < 7) {
        // Layer 1: buf0 -> buf1
        dense_layer(buf0, W1, b1, buf1, wave, lane);
    } else {
        // Wave 7: closed-form CIN + linear term, concurrent with layer 1.
        // p_plus @ W_cin == C0*S2[b] + C1*S3[b] + C2*S4[b] with Sk = sum_d x0_last^k.
        float t0 = 0.f, t1 = 0.f, t2 = 0.f;
        for (int h = lane; h < H_; h += 32) {
            t0 += cin_w0[h * (M_ * D_) + (M_ - 1) * D_ + (D_ - 1)] * W_cin[h];
            t1 += cin_w[h * (H_ * D_) + (H_ - 1) * D_ + (D_ - 1)] * W_cin[H_ + h];
            t2 += cin_w[(H_ + h) * (H_ * D_) + (H_ - 1) * D_ + (D_ - 1)] * W_cin[2 * H_ + h];
        }
        for (int off = 16; off >= 1; off >>= 1) {
            t0 += __shfl_down(t0, off, 32);
            t1 += __shfl_down(t1, off, 32);
            t2 += __shfl_down(t2, off, 32);
        }
        float C0 = __shfl(t0, 0, 32);
        float C1 = __shfl(t1, 0, 32);
        float C2 = __shfl(t2, 0, 32);
        const float w0_99 = cin_w0[(H_ - 1) * (M_ * D_) + (M_ - 1) * D_ + (D_ - 1)];
        const float w1_99 = cin_w[(H_ - 1) * (H_ * D_) + (H_ - 1) * D_ + (D_ - 1)];
        C1 *= w0_99;
        C2 *= w0_99 * w1_99;
        if (lane < B_) {
            float S2 = 0.f, S3 = 0.f, S4 = 0.f;
            for (int d = 0; d < D_; ++d) {
                float vv = buf0[lane][(M_ - 1) * D_ + d];  // x0_last[b, d]
                float v2 = vv * vv;
                S2 += v2; S3 += v2 * vv; S4 += v2 * v2;
            }
            float lin = 0.f;
            for (int m = 0; m < M_; ++m)
                lin += W_linear[m * V_ + x[lane * M_ + m]];
            base[lane] = lin + bias[0] + C0 * S2 + C1 * S3 + C2 * S4;
        }
    }
    __syncthreads();

    if (wave < 7) dense_layer(buf1, W2, b2, buf0, wave, lane);  // Layer 2
    __syncthreads();

    if (wave < 7) dense_layer(buf0, W3, b3, buf1, wave, lane);  // Layer 3 -> x_dnn
    __syncthreads();

    // Final: out[b] = sigmoid( x_dnn[b,:] . W_dnn + base[b] )
    if (tid < B_) {
        float acc = 0.f;
        for (int j = 0; j < NH; ++j) acc += buf1[tid][j] * W_dnn[j];
        float z = acc + base[tid];
        out[tid] = 1.0f / (1.0f + __expf(-z));
    }
}

extern "C" void kernel_launch(void* const* d_in, const int* in_sizes, int n_in,
                              void* d_out, int out_size, void* d_ws, size_t ws_size,
                              hipStream_t stream)
{
    (void)in_sizes; (void)n_in; (void)out_size; (void)d_ws; (void)ws_size;
    xdeepfm_fused<<<1, 256, 0, stream>>>(
        (const int*)  d_in[0],   // x
        (const float*)d_in[1],   // emb
        (const float*)d_in[2],   // cin_w0
        (const float*)d_in[3],   // cin_w
        (const float*)d_in[4],   // W_linear
        (const float*)d_in[5],   // W_dnn
        (const float*)d_in[6],   // W_cin
        (const float*)d_in[7],   // bias
        (const float*)d_in[8],   // W1
        (const float*)d_in[9],   // b1
        (const float*)d_in[10],  // W2
        (const float*)d_in[11],  // b2
        (const float*)d_in[12],  // W3
        (const float*)d_in[13],  // b3
        (float*)d_out);
}